// FlashMemory_44530220925620
// MI455X (gfx1250) — compile-verified
//
#include <hip/hip_runtime.h>
#include <hip/hip_bf16.h>
#include <math.h>

// ---------------- problem constants ----------------
#define XDIM      1176
#define TFRAMES   256
#define PERF      64          // 64 sub-blocks per frame in xs
#define LDK       75264       // PERF * XDIM, feature dim of xs/cent rows
#define KCLUST    60
#define KPAD      64
#define KM_ITERS  10
#define SPAT      30
#define FRAME_ELEMS 301056    // 256 * 1176, one frame of original x

#define KCHUNK    768         // K chunk per block (75264 = 768*98)
#define KSPLIT    98
#define KSTEPS    (KCHUNK/4)

// out layout (floats): spa_x [30*301056] | tem_x [60*75264] | idx(30 int) | ts(60 int) | w(60 f32)
#define TEM_OUT   9031680
#define IDX_OUT   13547520
#define TS_OUT    13547550
#define W_OUT     13547610

typedef __attribute__((ext_vector_type(2))) float v2f;
typedef __attribute__((ext_vector_type(8))) float v8f;

// ---------------- 1) temporal pool: x (65536x1176) -> xs (256x75264) ----------------
__global__ void pool_kernel(const float* __restrict__ x, float* __restrict__ xs) {
  int o = blockIdx.x * 256 + threadIdx.x;           // 19,267,584 outputs
  int e  = o % XDIM;
  int rb = o / XDIM;                                 // (t, nhh, nww, blk)
  int blk = rb & 3, ii = blk >> 1, jj = blk & 1;
  int nww = (rb >> 2) & 3;
  int nhh = (rb >> 4) & 3;
  int t   = rb >> 6;
  int v = e % 14; int tmp = e / 14; int u = tmp % 14; int cd = tmp / 14; // cd=c*2+d
  int a = u / 7, b = v / 7;
  int p0 = (u % 7) * 2, q0 = (v % 7) * 2;
  int r = ((t * 8 + nhh * 2 + ii) * 8 + nww * 2 + jj) * 4 + a * 2 + b;
  const float* src = x + (size_t)r * XDIM + (cd * 14 + p0) * 14 + q0;
  xs[o] = 0.25f * (src[0] + src[1] + src[14] + src[15]);
}

// ---------------- row sum-of-squares (deterministic tree) ----------------
__global__ void rownorm_kernel(const float* __restrict__ X, float* __restrict__ out) {
  __shared__ float red[256];
  int row = blockIdx.x;
  float s = 0.0f;
  for (int c = threadIdx.x; c < LDK; c += 256) {
    float v = X[(size_t)row * LDK + c];
    s += v * v;
  }
  red[threadIdx.x] = s; __syncthreads();
  for (int w = 128; w > 0; w >>= 1) {
    if (threadIdx.x < w) red[threadIdx.x] += red[threadIdx.x + w];
    __syncthreads();
  }
  if (threadIdx.x == 0) out[row] = red[0];
}

// ---------------- centroid init: cent[k] = xs[round(linspace(0,255,60))[k]] ----------------
__global__ void init_cent_kernel(const float* __restrict__ xs, float* __restrict__ cent) {
  int k = blockIdx.x;
  int col = blockIdx.y * 256 + threadIdx.x;
  int src = (int)rintf((float)k * (255.0f / 59.0f));   // round-half-even like jnp.round
  cent[(size_t)k * LDK + col] = xs[(size_t)src * LDK + col];
}

// ---------------- WMMA fp32 matmul: Gpart[kslice][m][n] = A[m,:]·B[n,:] over K chunk ----------------
// A rows optionally indirected through Aidx. One 16x16 tile per wave (V_WMMA_F32_16X16X4_F32).
// A tile is staged in LDS via the CDNA5 async memory->LDS path (ASYNCcnt).
__global__ void wmma_dot_kernel(const float* __restrict__ A, const int* __restrict__ Aidx,
                                const float* __restrict__ B, float* __restrict__ Gpart,
                                int ldg, int mtotal) {
  __shared__ __attribute__((aligned(16))) float As[16 * KCHUNK];   // 48 KB
  const int tid  = threadIdx.x;
  const int lane = tid & 31;
  const int warp = tid >> 5;
  const int m0 = blockIdx.x * 16;
  const int k0 = blockIdx.y * KCHUNK;

  // Async A-tile copy: 16 rows x KCHUNK floats = 3072 x b128 transfers, memory -> LDS
  // directly (no VGPR round-trip). All lanes active on every issue (exact multiples).
  for (int i = tid; i < 16 * (KCHUNK / 4); i += blockDim.x) {
    int r  = i / (KCHUNK / 4);
    int c4 = i % (KCHUNK / 4);
    int row = Aidx ? Aidx[m0 + r] : (m0 + r);
    const float* gsrc = A + (size_t)row * LDK + k0 + c4 * 4;
    unsigned ldsoff = (unsigned)(uintptr_t)(const void*)&As[r * KCHUNK + c4 * 4];
    unsigned long long gaddr = (unsigned long long)(uintptr_t)gsrc;
    asm volatile("global_load_async_to_lds_b128 %0, %1, off"
                 :: "v"(ldsoff), "v"(gaddr)
                 : "memory");
  }
  asm volatile("s_wait_asynccnt 0x0" ::: "memory");
  __syncthreads();

  const int n0    = warp * 16;
  const int arow  = lane & 15;
  const int apair = lane >> 4;                        // 0 or 1
  const int brow  = n0 + (lane & 15);
  const float4* B4   = (const float4*)(B + (size_t)brow * LDK + k0);  // 16B aligned
  const float*  Arow = &As[arow * KCHUNK + 2 * apair];

  v8f acc = {};
  for (int kk = 0; kk < KSTEPS; ++kk) {
    v2f a = *(const v2f*)(Arow + 4 * kk);             // A 16x4 layout: lane=M, pairs of K
    float4 q = B4[kk];                                // one b128 per k-step per lane
    v2f b;                                            // lane needs {k+apair, k+apair+2}
    b.x = apair ? q.y : q.x;
    b.y = apair ? q.w : q.z;
    acc = __builtin_amdgcn_wmma_f32_16x16x4_f32(
        /*neg_a=*/false, a, /*neg_b=*/false, b,
        /*c_mod=*/(short)0, acc, /*reuse_a=*/false, /*reuse_b=*/false);
  }

  float* Gp = Gpart + (size_t)blockIdx.y * ((size_t)mtotal * ldg);
  int mbase = m0 + 8 * apair;
  int ncol  = n0 + (lane & 15);
#pragma unroll
  for (int r = 0; r < 8; ++r) {
    Gp[(size_t)(mbase + r) * ldg + ncol] = acc[r];
  }
}

// fixed-order reduction of K-slice partials (deterministic)
__global__ void reduce_parts_kernel(const float* __restrict__ parts, float* __restrict__ out, int n) {
  int i = blockIdx.x * blockDim.x + threadIdx.x;
  if (i >= n) return;
  float s = 0.0f;
  for (int p = 0; p < KSPLIT; ++p) s += parts[(size_t)p * n + i];
  out[i] = s;
}

// ---------------- assignment + deterministic member lists ----------------
__global__ void assign_kernel(const float* __restrict__ G, const float* __restrict__ B2,
                              int* __restrict__ assign, int* __restrict__ counts,
                              int* __restrict__ members) {
  int f = threadIdx.x;                                // 256 threads == 256 frames
  if (f < KPAD) counts[f] = 0;
  __syncthreads();
  float best = INFINITY; int bi = 0;
  for (int k = 0; k < KCLUST; ++k) {
    float d = B2[k] - 2.0f * G[f * KPAD + k];         // A2[f] constant per row: argmin invariant
    if (d < best) { best = d; bi = k; }               // strict < keeps first (jnp.argmin)
  }
  assign[f] = bi;
  atomicAdd(&counts[bi], 1);
  __syncthreads();
  int slot = 0;                                       // rank among earlier frames -> deterministic
  for (int ff = 0; ff < f; ++ff) if (assign[ff] == bi) slot++;
  members[bi * TFRAMES + slot] = f;
}

// ---------------- centroid update: grouped row-sum (one-hot matmul) ----------------
__global__ void update_cent_kernel(const float* __restrict__ xs, const int* __restrict__ counts,
                                   const int* __restrict__ members, float* __restrict__ cent) {
  int c   = blockIdx.x;
  int col = blockIdx.y * 256 + threadIdx.x;
  int cnt = counts[c];
  if (cnt <= 0) return;                               // counts==0 -> keep old centroid
  float s = 0.0f;
  for (int i = 0; i < cnt; ++i)                       // ascending frame order -> deterministic
    s += xs[(size_t)members[c * TFRAMES + i] * LDK + col];
  cent[(size_t)c * LDK + col] = s / (float)cnt;
}

// ---------------- ordering, weights, timestamps, selection tables ----------------
__global__ void finalize_small_kernel(const int* __restrict__ assign, const int* __restrict__ counts,
                                      float* __restrict__ dout, int* __restrict__ selT,
                                      int* __restrict__ selS) {
  if (threadIdx.x != 0 || blockIdx.x != 0) return;
  float tsum[KCLUST]; float times[KCLUST];
  for (int c = 0; c < KCLUST; ++c) tsum[c] = 0.0f;
  for (int f = 0; f < TFRAMES; ++f) tsum[assign[f]] += (float)f;
  for (int c = 0; c < KCLUST; ++c)
    times[c] = tsum[c] / fmaxf((float)counts[c], 1.0f);

  // stable argsort ascending by times (insertion sort, shifts only on strict >)
  int ord[KCLUST];
  for (int i = 0; i < KCLUST; ++i) ord[i] = i;
  for (int i = 1; i < KCLUST; ++i) {
    int key = ord[i]; float kv = times[key]; int j = i - 1;
    while (j >= 0 && times[ord[j]] > kv) { ord[j + 1] = ord[j]; j--; }
    ord[j + 1] = key;
  }
  float w[KCLUST];
  int* ts_out = (int*)(dout + TS_OUT);
  for (int k = 0; k < KCLUST; ++k) {
    selT[k] = ord[k];                                 // tem_x row -> original centroid row
    w[k] = (float)counts[ord[k]];
    dout[W_OUT + k] = w[k];
    ts_out[k] = (int)rintf(times[ord[k]]);            // jnp.round: half-to-even
  }
  // stable argsort by -w (descending weight, ties keep ascending index)
  int ord2[KCLUST];
  for (int i = 0; i < KCLUST; ++i) ord2[i] = i;
  for (int i = 1; i < KCLUST; ++i) {
    int key = ord2[i]; float kv = w[key]; int j = i - 1;
    while (j >= 0 && w[ord2[j]] < kv) { ord2[j + 1] = ord2[j]; j--; }
    ord2[j + 1] = key;
  }
  for (int c = 0; c < SPAT; ++c) selS[c] = ord[ord2[c]];  // original centroid rows
  selS[30] = 0; selS[31] = 0;                             // pad rows for the 32-row WMMA tile
}

// ---------------- spatial idx: argmin_f sqrt(max(||c-x_f||^2,0)) == argmin_f (A2[f]-2H[c][f]) ----
__global__ void idx_kernel(const float* __restrict__ H, const float* __restrict__ A2,
                           int* __restrict__ idxWs, float* __restrict__ dout) {
  int c = threadIdx.x;
  if (c >= SPAT) return;
  float best = INFINITY; int bi = 0;
  for (int f = 0; f < TFRAMES; ++f) {
    float d = A2[f] - 2.0f * H[c * TFRAMES + f];
    if (d < best) { best = d; bi = f; }
  }
  idxWs[c] = bi;
  ((int*)dout)[IDX_OUT + c] = bi;
}

// ---------------- gathers into d_out ----------------
__global__ void spa_gather_kernel(const float* __restrict__ x, const int* __restrict__ idxWs,
                                  float* __restrict__ out) {
  int c = blockIdx.x;
  int off = blockIdx.y * 1024 + threadIdx.x * 4;      // 294*1024 = 301056
  const float4* src = (const float4*)(x + (size_t)idxWs[c] * FRAME_ELEMS + off);
  float4* dst = (float4*)(out + (size_t)c * FRAME_ELEMS + off);
  *dst = *src;
}

__global__ void tem_write_kernel(const float* __restrict__ cent, const int* __restrict__ selT,
                                 float* __restrict__ out) {
  int k = blockIdx.x;
  int col = blockIdx.y * 256 + threadIdx.x;
  out[TEM_OUT + (size_t)k * LDK + col] = cent[(size_t)selT[k] * LDK + col];
}

// ---------------- launcher ----------------
extern "C" void kernel_launch(void* const* d_in, const int* in_sizes, int n_in,
                              void* d_out, int out_size, void* d_ws, size_t ws_size,
                              hipStream_t stream) {
  (void)in_sizes; (void)n_in; (void)out_size; (void)ws_size;
  const float* x = (const float*)d_in[0];             // (65536, 1176) fp32
  float* out = (float*)d_out;
  char* ws = (char*)d_ws;

  // workspace layout (bytes)
  float* xs     = (float*)(ws + 0);                                  // 256*75264 f32
  float* cent   = (float*)(ws + 77070336);                           // 64*75264 f32 (60 used)
  float* Gpart  = (float*)(ws + 96337920);                           // 98*256*64 f32
  float* Hpart  = Gpart;                                             // reuse (sequential)
  float* G      = (float*)(ws + 102760448);                          // 256*64
  float* H      = (float*)(ws + 102825984);                          // 32*256
  float* A2     = (float*)(ws + 102858752);                          // 256
  float* B2     = (float*)(ws + 102859776);                          // 64
  int*   assign = (int*)  (ws + 102860032);                          // 256
  int*   counts = (int*)  (ws + 102861056);                          // 64
  int*   members= (int*)  (ws + 102861312);                          // 64*256
  int*   selT   = (int*)  (ws + 102926848);                          // 64
  int*   selS   = (int*)  (ws + 102927104);                          // 32
  int*   idxWs  = (int*)  (ws + 102927232);                          // 32

  // 1) pool (HBM-bound: 308MB read, 77MB write)
  pool_kernel<<<LDK, 256, 0, stream>>>(x, xs);
  // 2) frame norms (xs is L2-resident from here on)
  rownorm_kernel<<<TFRAMES, 256, 0, stream>>>(xs, A2);
  // 3) centroid init
  init_cent_kernel<<<dim3(KCLUST, 294), 256, 0, stream>>>(xs, cent);

  // 4) k-means: 10 x { G = xs@cent^T (WMMA f32), assign, grouped-sum update }
  for (int it = 0; it < KM_ITERS; ++it) {
    wmma_dot_kernel<<<dim3(16, KSPLIT), 128, 0, stream>>>(xs, nullptr, cent, Gpart, KPAD, TFRAMES);
    reduce_parts_kernel<<<(TFRAMES * KPAD + 255) / 256, 256, 0, stream>>>(Gpart, G, TFRAMES * KPAD);
    rownorm_kernel<<<KCLUST, 256, 0, stream>>>(cent, B2);
    assign_kernel<<<1, 256, 0, stream>>>(G, B2, assign, counts, members);
    update_cent_kernel<<<dim3(KCLUST, 294), 256, 0, stream>>>(xs, counts, members, cent);
  }
  // 5) final assignment with converged centroids
  wmma_dot_kernel<<<dim3(16, KSPLIT), 128, 0, stream>>>(xs, nullptr, cent, Gpart, KPAD, TFRAMES);
  reduce_parts_kernel<<<(TFRAMES * KPAD + 255) / 256, 256, 0, stream>>>(Gpart, G, TFRAMES * KPAD);
  rownorm_kernel<<<KCLUST, 256, 0, stream>>>(cent, B2);
  assign_kernel<<<1, 256, 0, stream>>>(G, B2, assign, counts, members);

  // 6) ordering, weights, timestamps, selection tables
  finalize_small_kernel<<<1, 32, 0, stream>>>(assign, counts, out, selT, selS);

  // 7) H = centroids(selS) @ xs^T  (32x256, WMMA f32), then idx = argmin per centroid
  wmma_dot_kernel<<<dim3(2, KSPLIT), 512, 0, stream>>>(cent, selS, xs, Hpart, TFRAMES, 32);
  reduce_parts_kernel<<<(32 * TFRAMES + 255) / 256, 256, 0, stream>>>(Hpart, H, 32 * TFRAMES);
  idx_kernel<<<1, 32, 0, stream>>>(H, A2, idxWs, out);

  // 8) outputs: spa_x gather from original x, tem_x from ordered centroids
  spa_gather_kernel<<<dim3(SPAT, 294), 256, 0, stream>>>(x, idxWs, out);
  tem_write_kernel<<<dim3(KCLUST, 294), 256, 0, stream>>>(cent, selT, out);
}